// GINLayer_27367531610540
// MI455X (gfx1250) — compile-verified
//
#include <hip/hip_runtime.h>

typedef __attribute__((ext_vector_type(2))) float v2f;
typedef __attribute__((ext_vector_type(8))) float v8f;

#define NNODES 10000
#define NEDGES 640000
#define DIN    128
#define DOUT   256
#define BN_EPS 1e-5f

// ---------------------------------------------------------------------------
// Zero-fill (agg buffer + BN stat accumulators must be zeroed every call)
// ---------------------------------------------------------------------------
__global__ void zero_f32(float* __restrict__ p, int n) {
    int i = blockIdx.x * blockDim.x + threadIdx.x;
    if (i < n) p[i] = 0.0f;
}

// ---------------------------------------------------------------------------
// Pack W[K,Nout] into pair-interleaved float2: Wp[p*Nout+n] = {W[2p][n], W[2p+1][n]}
// Lets each lane fetch its B-fragment K-pair with a single global_load_b64.
// ---------------------------------------------------------------------------
__global__ void pack_w_pairs(const float* __restrict__ W,
                             float2* __restrict__ Wp, int K, int Nout) {
    int i = blockIdx.x * blockDim.x + threadIdx.x;
    int total = (K >> 1) * Nout;
    if (i >= total) return;
    int p = i / Nout;
    int n = i - p * Nout;
    float2 v;
    v.x = W[(size_t)(2 * p) * Nout + n];
    v.y = W[(size_t)(2 * p + 1) * Nout + n];
    Wp[i] = v;
}

// ---------------------------------------------------------------------------
// Edge scatter-add: one wave32 per edge; lane l moves features [4l, 4l+4).
// x rows are 5.1MB total -> gathers hit L2; agg is L2-resident for atomics.
// ---------------------------------------------------------------------------
__global__ void scatter_add_edges(const float* __restrict__ x,
                                  const long long* __restrict__ ei,
                                  float* __restrict__ agg) {
    int gid  = blockIdx.x * blockDim.x + threadIdx.x;
    int lane = gid & 31;
    int edge = gid >> 5;
    if (edge >= NEDGES) return;
    long long s = ei[edge];            // row 0 of (2,E): src
    long long d = ei[NEDGES + edge];   // row 1 of (2,E): dst
    const float4* xr = (const float4*)(x + (size_t)s * DIN);
    float4 v = xr[lane];               // 32 lanes * 4 floats == DIN
    float* ar = agg + (size_t)d * DIN + lane * 4;
    atomicAdd(ar + 0, v.x);
    atomicAdd(ar + 1, v.y);
    atomicAdd(ar + 2, v.z);
    atomicAdd(ar + 3, v.w);
}

// ---------------------------------------------------------------------------
// f32 WMMA GEMM + bias:  Out[M,DOUT] = (A0 [+ A1]) @ W[K,DOUT] + bias
// One wave computes a 16x16 tile via V_WMMA_F32_16X16X4_F32 over K/4 steps.
// Block = 256 threads = 8 waves; block owns one 16-row stripe, waves stride
// over the DOUT/16 column tiles. M must be a multiple of 16 (10000 = 625*16).
//
// ISA 32-bit fragment layouts (CDNA5 ISA §7.12.2):
//   A 16x4 : lanes 0-15 -> M=lane, K={k0,k0+1}; lanes 16-31 -> M=lane-16, K={k0+2,k0+3}
//   B 4x16 : lanes 0-15 -> N=lane, K={k0,k0+1}; lanes 16-31 -> N=lane-16, K={k0+2,k0+3}
//   C/D    : VGPR r -> M=r (lanes 0-15) / M=r+8 (lanes 16-31), N=lane&15
//
// FUSE is a compile-time flag (GIN fuse: A = x + agg) so there is no runtime
// branch in the inner loop (keeps EXEC untouched around v_wmma).
// ---------------------------------------------------------------------------
template <int K, bool FUSE>
__global__ void gemm_bias_wmma_f32(const float* __restrict__ A0,
                                   const float* __restrict__ A1,
                                   const float2* __restrict__ Wp,
                                   const float* __restrict__ bias,
                                   float* __restrict__ Out) {
    const int wave  = threadIdx.x >> 5;
    const int lane  = threadIdx.x & 31;
    const int half  = lane >> 4;       // 0: lanes 0-15, 1: lanes 16-31
    const int l16   = lane & 15;
    const int mtile = blockIdx.x;

    const int rowA = mtile * 16 + l16;
    const float2* a0p = (const float2*)(A0 + (size_t)rowA * K);
    const float2* a1p = (const float2*)(A1 + (size_t)rowA * K);

    for (int ct = wave; ct < (DOUT >> 4); ct += 8) {
        const int n0 = ct * 16;
        const float2* bp = Wp + n0 + l16;
        v8f c = {};
        #pragma unroll 8
        for (int k0 = 0; k0 < K; k0 += 4) {
            const int kp = (k0 >> 1) + half;   // this lane's K-pair index
            float2 av = a0p[kp];               // global_load_b64
            if (FUSE) {
                float2 t = a1p[kp];            // global_load_b64 (v_pk_add_f32)
                av.x += t.x;
                av.y += t.y;
            }
            float2 bv = bp[(size_t)kp * DOUT]; // global_load_b64 (packed W)
            v2f a, b;
            a.x = av.x; a.y = av.y;
            b.x = bv.x; b.y = bv.y;
            // 8 args: (neg_a, A, neg_b, B, c_mod, C, reuse_a, reuse_b)
            c = __builtin_amdgcn_wmma_f32_16x16x4_f32(
                    false, a, false, b, (short)0, c, false, false);
        }
        const float bb = bias[n0 + l16];
        #pragma unroll
        for (int r = 0; r < 8; ++r) {
            const int mm = r + half * 8;
            Out[(size_t)(mtile * 16 + mm) * DOUT + n0 + l16] = c[r] + bb;
        }
    }
}

// ---------------------------------------------------------------------------
// BatchNorm column statistics: block = 256 threads (one per feature),
// each block reduces a stripe of rows (coalesced reads), one atomicAdd
// per feature per block into global sum / sumsq.
// ---------------------------------------------------------------------------
__global__ void bn_colstats(const float* __restrict__ H,
                            float* __restrict__ sum,
                            float* __restrict__ sumsq,
                            int rows, int rowsPerBlock) {
    int f  = threadIdx.x;                       // DOUT threads
    int r0 = blockIdx.x * rowsPerBlock;
    int r1 = r0 + rowsPerBlock;
    if (r1 > rows) r1 = rows;
    float s = 0.0f, q = 0.0f;
    for (int r = r0; r < r1; ++r) {
        float v = H[(size_t)r * DOUT + f];
        s += v;
        q += v * v;
    }
    atomicAdd(&sum[f], s);
    atomicAdd(&sumsq[f], q);
}

// ---------------------------------------------------------------------------
// BN normalize (training mode, biased variance) + ReLU, in place.
// ---------------------------------------------------------------------------
__global__ void bn_relu_apply(float* __restrict__ H,
                              const float* __restrict__ sum,
                              const float* __restrict__ sumsq,
                              const float* __restrict__ gamma,
                              const float* __restrict__ beta,
                              int rows) {
    int idx = blockIdx.x * blockDim.x + threadIdx.x;
    if (idx >= rows * DOUT) return;
    int f = idx & (DOUT - 1);
    float inv_n = 1.0f / (float)rows;
    float mean  = sum[f] * inv_n;
    float var   = sumsq[f] * inv_n - mean * mean;
    float is    = rsqrtf(var + BN_EPS);
    float v     = gamma[f] * (H[idx] - mean) * is + beta[f];
    H[idx] = v > 0.0f ? v : 0.0f;
}

// ---------------------------------------------------------------------------
extern "C" void kernel_launch(void* const* d_in, const int* in_sizes, int n_in,
                              void* d_out, int out_size, void* d_ws, size_t ws_size,
                              hipStream_t stream) {
    const float*     x   = (const float*)d_in[0];
    const long long* ei  = (const long long*)d_in[1];   // int64 (2,E)
    const float*     W1  = (const float*)d_in[2];
    const float*     b1  = (const float*)d_in[3];
    const float*     g1  = (const float*)d_in[4];
    const float*     be1 = (const float*)d_in[5];
    const float*     W2  = (const float*)d_in[6];
    const float*     b2  = (const float*)d_in[7];
    const float*     g2  = (const float*)d_in[8];
    const float*     be2 = (const float*)d_in[9];
    float*           out = (float*)d_out;

    // Workspace: agg[N*DIN] | h1[N*DOUT] | stats[4*DOUT] | Wp1[DIN/2*DOUT] | Wp2[DOUT/2*DOUT]
    float*  agg   = (float*)d_ws;
    float*  h1    = agg + (size_t)NNODES * DIN;
    float*  stats = h1 + (size_t)NNODES * DOUT;
    float*  sum1  = stats;
    float*  sq1   = stats + DOUT;
    float*  sum2  = stats + 2 * DOUT;
    float*  sq2   = stats + 3 * DOUT;
    float2* Wp1   = (float2*)(stats + 4 * DOUT);
    float2* Wp2   = Wp1 + (size_t)(DIN / 2) * DOUT;

    // 1) zero accumulators (required every call for deterministic replay)
    {
        int zn = NNODES * DIN;
        zero_f32<<<(zn + 255) / 256, 256, 0, stream>>>(agg, zn);
        zero_f32<<<(4 * DOUT + 255) / 256, 256, 0, stream>>>(stats, 4 * DOUT);
    }

    // 2) pack weights into pair-interleaved form (tiny: 128KB + 256KB)
    {
        int t1 = (DIN / 2) * DOUT;
        int t2 = (DOUT / 2) * DOUT;
        pack_w_pairs<<<(t1 + 255) / 256, 256, 0, stream>>>(W1, Wp1, DIN, DOUT);
        pack_w_pairs<<<(t2 + 255) / 256, 256, 0, stream>>>(W2, Wp2, DOUT, DOUT);
    }

    // 3) edge scatter-add (one wave32 per edge)
    {
        long long threads = (long long)NEDGES * 32;
        int blocks = (int)((threads + 255) / 256);
        scatter_add_edges<<<blocks, 256, 0, stream>>>(x, ei, agg);
    }

    // 4) GEMM1 (fused x + agg) + bias -> h1         [10000,128] @ [128,256]
    gemm_bias_wmma_f32<DIN, true><<<NNODES / 16, 256, 0, stream>>>(
        x, agg, Wp1, b1, h1);

    // 5) BN1 stats + apply + ReLU (in place on h1)
    bn_colstats<<<100, DOUT, 0, stream>>>(h1, sum1, sq1, NNODES, 100);
    bn_relu_apply<<<(NNODES * DOUT + 255) / 256, 256, 0, stream>>>(
        h1, sum1, sq1, g1, be1, NNODES);

    // 6) GEMM2 + bias -> d_out                      [10000,256] @ [256,256]
    gemm_bias_wmma_f32<DOUT, false><<<NNODES / 16, 256, 0, stream>>>(
        h1, h1 /*unused*/, Wp2, b2, out);

    // 7) BN2 stats + apply + ReLU (in place on d_out)
    bn_colstats<<<100, DOUT, 0, stream>>>(out, sum2, sq2, NNODES, 100);
    bn_relu_apply<<<(NNODES * DOUT + 255) / 256, 256, 0, stream>>>(
        out, sum2, sq2, g2, be2, NNODES);
}